// ScaledDotProduct_66486093742705
// MI455X (gfx1250) — compile-verified
//
#include <hip/hip_runtime.h>
#include <stdint.h>

// Problem constants (fixed by the reference).
#define NROWS 8192      // N: output rows, also key's column count
#define DCOLS 4096      // D: row length for softmax, also key's row count
#define TI    32        // rows per block  -> key loads are 128B-coalesced
#define CJ    64        // columns per chunk
#define NCH   (DCOLS / CJ)
#define PADW  68        // LDS row pitch (floats): %4==0 -> aligned b128 reads
#define RPAD  17        // reduction array pitch (floats), odd -> conflict-free
// SC2 = (1/sqrt(4096)) * log2(e): puts everything in the exp2 domain.
#define SC2   0.022542110f

typedef float v4f __attribute__((ext_vector_type(4)));

// ---- CDNA5 async global->LDS copy (ASYNCcnt path) -------------------------
// Per-lane: copies 4 bytes from global address gp to LDS byte address lds_byte_addr. The per-lane LDS
// destination lets us do the key transpose as part of the async DMA.
__device__ __forceinline__ void async_ld_b32(uint32_t lds_byte_addr,
                                             const float* gp) {
  asm volatile("global_load_async_to_lds_b32 %0, %1, off"
               :: "v"(lds_byte_addr), "v"((uint64_t)(uintptr_t)gp)
               : "memory");
}

template <int N>
__device__ __forceinline__ void wait_async_le() {
#if __has_builtin(__builtin_amdgcn_s_wait_asynccnt)
  __builtin_amdgcn_s_wait_asynccnt(N);
#else
  asm volatile("s_wait_asynccnt %0" :: "i"(N) : "memory");
#endif
}

__device__ __forceinline__ float exp2_fast(float x) {
#if __has_builtin(__builtin_amdgcn_exp2f)
  return __builtin_amdgcn_exp2f(x);
#else
  return exp2f(x);
#endif
}

__device__ __forceinline__ float log2_fast(float x) {
#if __has_builtin(__builtin_amdgcn_logf)
  return __builtin_amdgcn_logf(x);
#else
  return log2f(x);
#endif
}

__global__ __launch_bounds__(256)
void ScaledDotProduct_66486093742705_kernel(const float* __restrict__ q,
                                            const float* __restrict__ k,
                                            const float* __restrict__ v,
                                            float* __restrict__ out) {
  __shared__ float kT[2][TI][PADW]; // double-buffered transposed key tile
  __shared__ float Mred[TI][RPAD];  // per-row partial max (16 contributors)
  __shared__ float Sred[TI][RPAD];  // per-row partial rescaled sums
  __shared__ float crow[TI];        // per-row  m + log2(s)

  const int tid  = threadIdx.x;
  const int i0   = blockIdx.x * TI;
  // key staging mapping: lane -> (iiK, jsub); wave = 32 consecutive i = 128B
  const int iiK  = tid & 31;
  const int jsub = tid >> 5;          // 0..7 (wave id)
  // compute mapping: thread -> 4 consecutive cols jv, row-pair group irg
  const int c16  = tid & 15;          // 0..15
  const int jv   = c16 * 4;           // 0,4,...,60
  const int irg  = tid >> 4;          // 0..15 -> rows irg*2, irg*2+1

  const uint32_t ldsK0 = (uint32_t)(uintptr_t)(&kT[0][0][0]);
  const uint32_t ldsK1 = (uint32_t)(uintptr_t)(&kT[1][0][0]);
  // This thread's fixed LDS scatter offset within a buffer.
  const uint32_t ldsOff = (uint32_t)((iiK * PADW + jsub * 8) * 4);
  // This thread's fixed key offset (elements) within a chunk.
  const uint32_t kOff = (uint32_t)(jsub * 8) * NROWS + (uint32_t)(i0 + iiK);
  const float NEG_INF = -__builtin_inff();

  // Issue the 8 async b32 scatter-transposes for one key chunk.
  auto stage_key = [&](uint32_t ldsBase, int jc) {
    const float* kp = k + (uint32_t)jc * NROWS + kOff;
#pragma unroll
    for (int p = 0; p < 8; ++p)
      async_ld_b32(ldsBase + ldsOff + (uint32_t)(p * 4),
                   kp + (uint32_t)(p * NROWS));
  };

  float m2[2], s2[2];
#pragma unroll
  for (int p = 0; p < 2; ++p) { m2[p] = NEG_INF; s2[p] = 0.0f; }

  // ---------------- Pass 1: online softmax statistics (m, s) per row ------
  stage_key(ldsK0, 0);
  for (int c = 0; c < NCH; ++c) {
    const int jc = c * CJ;
    const int b  = c & 1;
    if (c + 1 < NCH) {
      stage_key(b ? ldsK0 : ldsK1, jc + CJ);  // prefetch next key tile
      wait_async_le<8>();                     // in-order: current tile landed
    } else {
      wait_async_le<0>();
    }
    __syncthreads();

#pragma unroll
    for (int p = 0; p < 2; ++p) {
      const int ii = irg * 2 + p;
      const uint32_t base = (uint32_t)((i0 + ii) * DCOLS + jc + jv);
      const v4f q4 = *(const v4f*)(q + base);    // global_load_b128
      const v4f v4 = *(const v4f*)(v + base);    // global_load_b128
      const v4f k4 = *(const v4f*)(&kT[b][ii][jv]); // ds_load_b128
      const v4f y4 = (q4 * k4) * v4 * SC2;       // base-2 domain
      const float cm = fmaxf(fmaxf(y4.x, y4.y), fmaxf(y4.z, y4.w));
      const float mn = fmaxf(m2[p], cm);
      const float r  = exp2_fast(m2[p] - mn);    // rescale old sum
      const float e0 = exp2_fast(y4.x - mn);     // 4 independent exp2s
      const float e1 = exp2_fast(y4.y - mn);
      const float e2 = exp2_fast(y4.z - mn);
      const float e3 = exp2_fast(y4.w - mn);
      s2[p] = __builtin_fmaf(s2[p], r, (e0 + e1) + (e2 + e3));
      m2[p] = mn;
      if (c + 1 < NCH) {                         // global_prefetch_b8
        __builtin_prefetch(q + base + CJ, 0, 0);
        __builtin_prefetch(v + base + CJ, 0, 0);
      }
    }
    __syncthreads();
  }

  // ---------------- Reduce (m, s) across the 16 column-threads per row ----
#pragma unroll
  for (int p = 0; p < 2; ++p) {
    const int ii = irg * 2 + p;
    Mred[ii][c16] = m2[p];
    Sred[ii][c16] = s2[p];
  }
  __syncthreads();
  if (tid < TI) {
    float m = NEG_INF;
    for (int c = 0; c < 16; ++c) m = fmaxf(m, Mred[tid][c]);
    float s = 0.0f;
    for (int c = 0; c < 16; ++c) s += Sred[tid][c] * exp2_fast(Mred[tid][c] - m);
    crow[tid] = m + log2_fast(s);   // out = exp2(y - crow)
  }
  __syncthreads();

  // Hoist this thread's per-row constants into registers.
  float cr[2];
#pragma unroll
  for (int p = 0; p < 2; ++p) cr[p] = crow[irg * 2 + p];

  // ---------------- Pass 2: recompute x (q/k/v hot in L2), write softmax --
  stage_key(ldsK0, 0);
  for (int c = 0; c < NCH; ++c) {
    const int jc = c * CJ;
    const int b  = c & 1;
    if (c + 1 < NCH) {
      stage_key(b ? ldsK0 : ldsK1, jc + CJ);
      wait_async_le<8>();
    } else {
      wait_async_le<0>();
    }
    __syncthreads();

#pragma unroll
    for (int p = 0; p < 2; ++p) {
      const int ii = irg * 2 + p;
      const uint32_t base = (uint32_t)((i0 + ii) * DCOLS + jc + jv);
      const v4f q4 = *(const v4f*)(q + base);
      const v4f v4 = *(const v4f*)(v + base);
      const v4f k4 = *(const v4f*)(&kT[b][ii][jv]);
      const v4f t4 = (q4 * k4) * v4;
      v4f o4;
      o4.x = exp2_fast(__builtin_fmaf(t4.x, SC2, -cr[p]));
      o4.y = exp2_fast(__builtin_fmaf(t4.y, SC2, -cr[p]));
      o4.z = exp2_fast(__builtin_fmaf(t4.z, SC2, -cr[p]));
      o4.w = exp2_fast(__builtin_fmaf(t4.w, SC2, -cr[p]));
      __builtin_nontemporal_store(o4, (v4f*)(out + base)); // b128 NT store
    }
    __syncthreads();
  }
}

extern "C" void kernel_launch(void* const* d_in, const int* in_sizes, int n_in,
                              void* d_out, int out_size, void* d_ws, size_t ws_size,
                              hipStream_t stream) {
  (void)in_sizes; (void)n_in; (void)out_size; (void)d_ws; (void)ws_size;
  const float* q = (const float*)d_in[0];  // query [N, D]
  const float* k = (const float*)d_in[1];  // key   [D, N]
  const float* v = (const float*)d_in[2];  // value [N, D]
  float* out = (float*)d_out;              // [N, D] f32

  dim3 grid(NROWS / TI);   // 256 blocks
  dim3 block(256);         // 8 wave32s
  hipLaunchKernelGGL(ScaledDotProduct_66486093742705_kernel,
                     grid, block, 0, stream, q, k, v, out);
}